// PiecewiseHawkesIntensity_13125420057297
// MI455X (gfx1250) — compile-verified
//
#include <hip/hip_runtime.h>
#include <cstdint>
#include <cstddef>

// Problem constants (from the reference): B=8, P=16, L=256, M=16, L_EVAL=2048
#define Bc 8
#define Pc 16
#define Lc 256
#define Mc 16
#define LEc 2048
#define CHUNK 256                 // queries per block
#define NCHUNK (LEc / CHUNK)      // 8 chunks -> grid = 8*16*8 = 1024 blocks

// gfx1250 async DMA: copy 16B per active lane from global memory directly into LDS.
// VDST operand = per-lane LDS byte offset (workgroup-relative; HW adds LDS_BASE),
// VADDR = per-lane 64-bit global address, saddr disabled ("off").
// Tracked by ASYNCcnt -> drained with s_wait_asynccnt.
__device__ __forceinline__ void async_copy_b128(void* lds_ptr, const void* gptr) {
  // Low 32 bits of a flat pointer into LDS are the workgroup-relative LDS offset.
  uint32_t lds_off = (uint32_t)(uintptr_t)lds_ptr;
  asm volatile("global_load_async_to_lds_b128 %0, %1, off"
               :
               : "v"(lds_off), "v"(gptr)
               : "memory");
}

__device__ __forceinline__ void wait_async_zero() {
  asm volatile("s_wait_asynccnt 0" ::: "memory");
}

__global__ __launch_bounds__(CHUNK) void hawkes_intensity_kernel(
    const float* __restrict__ query_times,  // [B,P,Le]
    const float* __restrict__ event_times,  // [B,P,L] (sorted along L)
    const float* __restrict__ mu,           // [B,M,P,L]
    const float* __restrict__ alpha,        // [B,M,P,L]
    const float* __restrict__ beta,         // [B,M,P,L]
    float* __restrict__ out)                // [B,M,P,Le]
{
  // LDS staging: events (1 KB) + mu/alpha/beta (3 * 16 KB) = 49 KB per block.
  __shared__ float smem[Lc + 3 * Mc * Lc];
  float* s_ev = smem;
  float* s_mu = smem + Lc;
  float* s_al = s_mu + Mc * Lc;
  float* s_be = s_al + Mc * Lc;

  const int tid   = threadIdx.x;
  const int blk   = blockIdx.x;
  const int chunk = blk % NCHUNK;
  const int p     = (blk / NCHUNK) % Pc;
  const int b     = blk / (NCHUNK * Pc);

  // ---- Stage event_times[b,p,:] : 64 x float4 via async DMA (waves 0-1) ----
  const float* evb = event_times + (b * Pc + p) * Lc;
  if (tid < Lc / 4) {
    async_copy_b128(&s_ev[tid * 4], evb + tid * 4);
  }

  // ---- Stage mu/alpha/beta: 16 rows x 256 floats each = 1024 float4 per array ----
  // Row m of array X lives at X[((b*M + m)*P + p)*L ...].
#pragma unroll
  for (int it = 0; it < (Mc * Lc / 4) / CHUNK; ++it) {   // 4 iterations
    const int f4 = it * CHUNK + tid;                     // 0..1023 (float4 index)
    const int m  = f4 >> 6;                              // row (64 float4 per row)
    const int c  = (f4 & 63) * 4;                        // float column in row
    const int g  = ((b * Mc + m) * Pc + p) * Lc + c;
    async_copy_b128(&s_mu[f4 * 4], mu    + g);
    async_copy_b128(&s_al[f4 * 4], alpha + g);
    async_copy_b128(&s_be[f4 * 4], beta  + g);
  }

  wait_async_zero();     // drain this wave's ASYNCcnt
  __syncthreads();       // make LDS writes visible across all 8 waves

  // ---- One query per thread ----
  const int   q = chunk * CHUNK + tid;
  const float t = query_times[(b * Pc + p) * LEc + q];

  // Branchless lower_bound over 256 sorted events in LDS: cnt = #{events < t}.
  int cnt = 0;
#pragma unroll
  for (int s = 128; s > 0; s >>= 1) {
    if (s_ev[cnt + s - 1] < t) cnt += s;
  }
  if (cnt == Lc - 1 && s_ev[Lc - 1] < t) cnt = Lc;  // all-256 case

  const int   ci     = (cnt == 0) ? 0 : (cnt - 1);  // clipped last index
  const float t_last = (cnt == 0) ? 0.0f : s_ev[ci];
  const float dt     = t - t_last;

  // Amortize the search across all M=16 parameter sets; gathers hit LDS.
  // softplus(x) = max(x,0) + log(1 + exp(-|x|)).
  // Here x = (1-e)*mu + e*alpha with mu,alpha in [0,1), e in (0,1], so
  // exp(-|x|) in (0.36, 1] -> no cancellation: native v_log_f32 is exact enough,
  // avoiding the heavy ocml log1p libcall in the 16x-unrolled hot loop.
  const int obase = ((b * Mc) * Pc + p) * LEc + q;
#pragma unroll
  for (int m = 0; m < Mc; ++m) {
    const float muv  = s_mu[m * Lc + ci];
    const float av   = s_al[m * Lc + ci];
    const float bv   = s_be[m * Lc + ci];
    const float e    = __expf(-bv * dt);
    const float base = fmaf(av - muv, e, muv);
    const float sp   = fmaxf(base, 0.0f) + __logf(1.0f + __expf(-fabsf(base)));
    out[obase + m * (Pc * LEc)] = sp;   // coalesced 128B/wave per m-slice
  }
}

extern "C" void kernel_launch(void* const* d_in, const int* in_sizes, int n_in,
                              void* d_out, int out_size, void* d_ws, size_t ws_size,
                              hipStream_t stream) {
  // setup_inputs() order: query_times, event_times, mu, alpha, beta (all fp32)
  const float* qt = (const float*)d_in[0];
  const float* ev = (const float*)d_in[1];
  const float* mu = (const float*)d_in[2];
  const float* al = (const float*)d_in[3];
  const float* be = (const float*)d_in[4];
  float* out = (float*)d_out;

  dim3 grid(Bc * Pc * NCHUNK);  // 1024 blocks
  dim3 block(CHUNK);            // 256 threads = 8 wave32
  hipLaunchKernelGGL(hawkes_intensity_kernel, grid, block, 0, stream,
                     qt, ev, mu, al, be, out);
}